// SFD_72430328481295
// MI455X (gfx1250) — compile-verified
//
#include <hip/hip_runtime.h>

#define BATCH    4
#define NPRI     34125
#define NSORT    65536          // power-of-two pad per batch for bitonic sort
#define KTOP     5000           // NMS_TOP_K
#define TOPK_OUT 750            // TOP_K
#define WPAD     160            // 32-bit words per mask row (>= ceil(5000/32)=157)
#define CONF_T   0.05f
#define NMS_T    0.3f

typedef unsigned long long u64;
typedef unsigned int       u32;

// ---------------------------------------------------------------- zero output
__global__ void zero_out_kernel(float* out, int n) {
    int t = blockIdx.x * blockDim.x + threadIdx.x;
    if (t < n) out[t] = 0.0f;
}

// ------------------------------------------- decode boxes + build sort keys
// key = (monotone-u32(score_or_-1e30) << 32) | (0xFFFFFFFF - prior_idx)
// descending u64 sort == JAX top_k order (desc score, asc index on ties).
__global__ void decode_keys_kernel(const float* __restrict__ loc,
                                   const float* __restrict__ conf,
                                   const float* __restrict__ prior,
                                   float* __restrict__ dec,
                                   u64* __restrict__ keys) {
    int t = blockIdx.x * blockDim.x + threadIdx.x;
    if (t >= BATCH * NSORT) return;
    int b = t >> 16;            // NSORT == 65536
    int p = t & (NSORT - 1);
    if (p < NPRI) {
        float p0 = prior[p * 4 + 0], p1 = prior[p * 4 + 1];
        float pw = prior[p * 4 + 2], ph = prior[p * 4 + 3];
        const float* l = loc + ((size_t)b * NPRI + p) * 4;
        float cx = __fadd_rn(p0, __fmul_rn(__fmul_rn(l[0], 0.1f), pw));
        float cy = __fadd_rn(p1, __fmul_rn(__fmul_rn(l[1], 0.1f), ph));
        float w  = __fmul_rn(pw, expf(__fmul_rn(l[2], 0.2f)));
        float h  = __fmul_rn(ph, expf(__fmul_rn(l[3], 0.2f)));
        float x1 = __fsub_rn(cx, __fmul_rn(w, 0.5f));
        float y1 = __fsub_rn(cy, __fmul_rn(h, 0.5f));
        float x2 = __fadd_rn(x1, w);
        float y2 = __fadd_rn(y1, h);
        float* d = dec + ((size_t)b * NPRI + p) * 4;
        d[0] = x1; d[1] = y1; d[2] = x2; d[3] = y2;
        float sc = conf[((size_t)b * NPRI + p) * 2 + 1];
        float s  = (sc > CONF_T) ? sc : -1e30f;
        u32 u = __float_as_uint(s);
        u = (u & 0x80000000u) ? ~u : (u | 0x80000000u);   // total-order flip
        keys[t] = ((u64)u << 32) | (u64)(0xFFFFFFFFu - (u32)p);
    } else {
        keys[t] = 0ull;         // sorts after every real entry (real u never 0)
    }
}

// ------------------------------------------------------- bitonic sort (desc)
__global__ __launch_bounds__(1024) void bitonic_local_full_kernel(u64* keys) {
    __shared__ u64 sh[2048];
    size_t base = (size_t)blockIdx.x * 2048;
    int t = threadIdx.x;
    sh[t] = keys[base + t];
    sh[t + 1024] = keys[base + t + 1024];
    __syncthreads();
    for (int k = 2; k <= 2048; k <<= 1) {
        for (int j = k >> 1; j >= 1; j >>= 1) {
            int i  = ((t & ~(j - 1)) << 1) | (t & (j - 1));
            int gi = (int)((base + (size_t)i) & (NSORT - 1));   // in-batch idx
            bool dir = ((gi & k) == 0);                         // descending blk
            u64 a = sh[i], c = sh[i | j];
            if ((a < c) == dir) { sh[i] = c; sh[i | j] = a; }
            __syncthreads();
        }
    }
    keys[base + t] = sh[t];
    keys[base + t + 1024] = sh[t + 1024];
}

__global__ void bitonic_global_kernel(u64* keys, int k, int j) {
    int t = blockIdx.x * blockDim.x + threadIdx.x;
    if (t >= BATCH * (NSORT / 2)) return;
    int b  = t / (NSORT / 2);
    int tt = t - b * (NSORT / 2);
    int i  = ((tt & ~(j - 1)) << 1) | (tt & (j - 1));
    u64* bp = keys + (size_t)b * NSORT;
    u64 a = bp[i], c = bp[i | j];
    bool dir = ((i & k) == 0);
    if ((a < c) == dir) { bp[i] = c; bp[i | j] = a; }
}

__global__ __launch_bounds__(1024) void bitonic_local_merge_kernel(u64* keys, int k) {
    __shared__ u64 sh[2048];
    size_t base = (size_t)blockIdx.x * 2048;
    int t = threadIdx.x;
    sh[t] = keys[base + t];
    sh[t + 1024] = keys[base + t + 1024];
    __syncthreads();
    for (int j = 1024; j >= 1; j >>= 1) {
        int i  = ((t & ~(j - 1)) << 1) | (t & (j - 1));
        int gi = (int)((base + (size_t)i) & (NSORT - 1));
        bool dir = ((gi & k) == 0);
        u64 a = sh[i], c = sh[i | j];
        if ((a < c) == dir) { sh[i] = c; sh[i | j] = a; }
        __syncthreads();
    }
    keys[base + t] = sh[t];
    keys[base + t + 1024] = sh[t + 1024];
}

// ---------------------------------------------------- gather top-K (SoA form)
__global__ void gather_topk_kernel(const u64* __restrict__ keys,
                                   const float* __restrict__ dec,
                                   const float* __restrict__ conf,
                                   float* __restrict__ X1, float* __restrict__ Y1,
                                   float* __restrict__ X2, float* __restrict__ Y2,
                                   float* __restrict__ AR, float* __restrict__ SC) {
    int t = blockIdx.x * blockDim.x + threadIdx.x;
    if (t >= BATCH * KTOP) return;
    int b = t / KTOP;
    int j = t - b * KTOP;
    u64 key = keys[(size_t)b * NSORT + j];
    u32 p = 0xFFFFFFFFu - (u32)(key & 0xFFFFFFFFull);
    if (p >= NPRI) p = NPRI - 1;   // never hit: >=5000 real entries always
    const float* d = dec + ((size_t)b * NPRI + p) * 4;
    float x1 = d[0], y1 = d[1], x2 = d[2], y2 = d[3];
    X1[t] = x1; Y1[t] = y1; X2[t] = x2; Y2[t] = y2;
    AR[t] = __fmul_rn(__fsub_rn(x2, x1), __fsub_rn(y2, y1));
    SC[t] = conf[((size_t)b * NPRI + p) * 2 + 1];
}

// ------------------------------------ pairwise IoU suppression-mask (bitmask)
// mask[b][i][w] bit t == ( iou(i, w*32+t) > NMS_T  &&  (w*32+t) > i )
// j-box tile staged into LDS with CDNA5 async global->LDS copies.
__global__ __launch_bounds__(256) void iou_mask_kernel(
        const float* __restrict__ X1, const float* __restrict__ Y1,
        const float* __restrict__ X2, const float* __restrict__ Y2,
        const float* __restrict__ AR, u32* __restrict__ mask) {
    __shared__ float sJ[5][256];
    __shared__ float sI[5][32];
    int b   = blockIdx.z;
    int i0  = blockIdx.x * 32;   // 32 i-rows per block
    int w0  = blockIdx.y * 8;    // 8 mask words (256 j-cols) per block
    int tid = threadIdx.x;

    {   // async-copy 256 j-boxes (5 SoA streams) into LDS
        int jj = w0 * 32 + tid;
        if (jj >= KTOP) jj = KTOP - 1;              // clamped dup, bits masked later
        size_t gj = (size_t)b * KTOP + jj;
        unsigned long long g0 = (unsigned long long)(X1 + gj);
        unsigned long long g1 = (unsigned long long)(Y1 + gj);
        unsigned long long g2 = (unsigned long long)(X2 + gj);
        unsigned long long g3 = (unsigned long long)(Y2 + gj);
        unsigned long long g4 = (unsigned long long)(AR + gj);
        unsigned l0 = (unsigned)(unsigned long long)&sJ[0][tid];
        unsigned l1 = (unsigned)(unsigned long long)&sJ[1][tid];
        unsigned l2 = (unsigned)(unsigned long long)&sJ[2][tid];
        unsigned l3 = (unsigned)(unsigned long long)&sJ[3][tid];
        unsigned l4 = (unsigned)(unsigned long long)&sJ[4][tid];
        asm volatile("global_load_async_to_lds_b32 %0, %1, off" :: "v"(l0), "v"(g0) : "memory");
        asm volatile("global_load_async_to_lds_b32 %0, %1, off" :: "v"(l1), "v"(g1) : "memory");
        asm volatile("global_load_async_to_lds_b32 %0, %1, off" :: "v"(l2), "v"(g2) : "memory");
        asm volatile("global_load_async_to_lds_b32 %0, %1, off" :: "v"(l3), "v"(g3) : "memory");
        asm volatile("global_load_async_to_lds_b32 %0, %1, off" :: "v"(l4), "v"(g4) : "memory");
    }
    if (tid < 32) {              // 32 i-boxes via normal loads
        int ii = i0 + tid;
        if (ii >= KTOP) ii = KTOP - 1;
        size_t gi = (size_t)b * KTOP + ii;
        sI[0][tid] = X1[gi]; sI[1][tid] = Y1[gi];
        sI[2][tid] = X2[gi]; sI[3][tid] = Y2[gi];
        sI[4][tid] = AR[gi];
    }
    asm volatile("s_wait_asynccnt 0x0" ::: "memory");
    __syncthreads();

    int li = tid >> 3;           // i-row within tile   (0..31)
    int lw = tid & 7;            // word within tile    (0..7)
    int i  = i0 + li;
    float x1i = sI[0][li], y1i = sI[1][li];
    float x2i = sI[2][li], y2i = sI[3][li], ai = sI[4][li];
    int w  = w0 + lw;
    int jb = lw * 32;
    u32 m = 0;
#pragma unroll 8
    for (int tj = 0; tj < 32; ++tj) {
        int lj = jb + tj;
        int j  = w * 32 + tj;
        float iw = fmaxf(__fsub_rn(fminf(x2i, sJ[2][lj]), fmaxf(x1i, sJ[0][lj])), 0.0f);
        float ih = fmaxf(__fsub_rn(fminf(y2i, sJ[3][lj]), fmaxf(y1i, sJ[1][lj])), 0.0f);
        float inter = __fmul_rn(iw, ih);
        float den   = __fsub_rn(__fadd_rn(ai, sJ[4][lj]), inter);
        float iou   = inter / den;
        bool cond = (iou > NMS_T) && (j > i) && (j < KTOP);
        m |= (cond ? 1u : 0u) << tj;
    }
    if (i < KTOP) mask[((size_t)b * KTOP + i) * WPAD + w] = m;
}

// --------------------------- serial greedy scan: one wave32 per batch image.
// Suppressed bitset: 160 words = lane-strided 5 VGPRs/lane across the wave.
__global__ __launch_bounds__(32) void nms_scan_kernel(
        const float* __restrict__ X1, const float* __restrict__ Y1,
        const float* __restrict__ X2, const float* __restrict__ Y2,
        const float* __restrict__ SC, const u32* __restrict__ mask,
        float* __restrict__ out) {
    int b    = blockIdx.x;
    int lane = threadIdx.x;
    u32 s0 = 0, s1 = 0, s2 = 0, s3 = 0, s4 = 0;
    int kept_count = 0;
    const u32*  mbase = mask + (size_t)b * KTOP * WPAD;
    const float* sc   = SC + (size_t)b * KTOP;
    for (int i = 0; i < KTOP; ++i) {
        int w     = i >> 5;
        int r     = w >> 5;
        int owner = w & 31;            // uniform
        u32 sw;
        switch (r) {
            case 0:  sw = s0; break;
            case 1:  sw = s1; break;
            case 2:  sw = s2; break;
            case 3:  sw = s3; break;
            default: sw = s4; break;
        }
        u32 word = (u32)__builtin_amdgcn_readlane((int)sw, owner);   // SGPR bcast
        float s = sc[i];
        bool kept = (s > CONF_T) && (((word >> (i & 31)) & 1u) == 0u);
        if (kept) {                    // wave-uniform branch
            const u32* row = mbase + (size_t)i * WPAD;
            s0 |= row[lane];
            s1 |= row[lane + 32];
            s2 |= row[lane + 64];
            s3 |= row[lane + 96];
            s4 |= row[lane + 128];
            if (kept_count < TOPK_OUT && lane < 5) {
                size_t gi = (size_t)b * KTOP + i;
                float v;
                if      (lane == 0) v = s;
                else if (lane == 1) v = X1[gi];
                else if (lane == 2) v = Y1[gi];
                else if (lane == 3) v = X2[gi];
                else                v = Y2[gi];
                out[(((size_t)b * 2 + 1) * TOPK_OUT + kept_count) * 5 + lane] = v;
            }
            kept_count++;
        }
    }
}

// ------------------------------------------------------------------- launcher
extern "C" void kernel_launch(void* const* d_in, const int* in_sizes, int n_in,
                              void* d_out, int out_size, void* d_ws, size_t ws_size,
                              hipStream_t stream) {
    (void)in_sizes; (void)n_in; (void)ws_size;
    const float* loc   = (const float*)d_in[0];
    const float* conf  = (const float*)d_in[1];
    const float* prior = (const float*)d_in[2];
    float* out = (float*)d_out;

    char* ws = (char*)d_ws;
    size_t off = 0;
    auto take = [&](size_t bytes) -> char* {
        char* p = ws + off;
        off += (bytes + 255) & ~(size_t)255;
        return p;
    };
    u64*   keys = (u64*)  take((size_t)BATCH * NSORT * sizeof(u64));      // 2.00 MB
    float* dec  = (float*)take((size_t)BATCH * NPRI * 4 * sizeof(float)); // 2.18 MB
    float* X1   = (float*)take((size_t)BATCH * KTOP * sizeof(float));
    float* Y1   = (float*)take((size_t)BATCH * KTOP * sizeof(float));
    float* X2   = (float*)take((size_t)BATCH * KTOP * sizeof(float));
    float* Y2   = (float*)take((size_t)BATCH * KTOP * sizeof(float));
    float* AR   = (float*)take((size_t)BATCH * KTOP * sizeof(float));
    float* SC   = (float*)take((size_t)BATCH * KTOP * sizeof(float));
    u32*   mask = (u32*)  take((size_t)BATCH * KTOP * WPAD * sizeof(u32)); // 12.8 MB

    zero_out_kernel<<<(out_size + 255) / 256, 256, 0, stream>>>(out, out_size);

    decode_keys_kernel<<<(BATCH * NSORT) / 256, 256, 0, stream>>>(loc, conf, prior, dec, keys);

    // Full descending bitonic sort of each batch's 65536 keys.
    bitonic_local_full_kernel<<<BATCH * NSORT / 2048, 1024, 0, stream>>>(keys);
    for (int k = 4096; k <= NSORT; k <<= 1) {
        for (int j = k >> 1; j >= 2048; j >>= 1)
            bitonic_global_kernel<<<BATCH * (NSORT / 2) / 256, 256, 0, stream>>>(keys, k, j);
        bitonic_local_merge_kernel<<<BATCH * NSORT / 2048, 1024, 0, stream>>>(keys, k);
    }

    gather_topk_kernel<<<(BATCH * KTOP + 255) / 256, 256, 0, stream>>>(
        keys, dec, conf, X1, Y1, X2, Y2, AR, SC);

    dim3 mg((KTOP + 31) / 32, WPAD / 8, BATCH);  // 157 x 20 x 4
    iou_mask_kernel<<<mg, 256, 0, stream>>>(X1, Y1, X2, Y2, AR, mask);

    nms_scan_kernel<<<BATCH, 32, 0, stream>>>(X1, Y1, X2, Y2, SC, mask, out);
}